// VQNCA_61297773248860
// MI455X (gfx1250) — compile-verified
//
#include <hip/hip_runtime.h>
#include <math.h>

typedef __attribute__((ext_vector_type(2))) float v2f;
typedef __attribute__((ext_vector_type(8))) float v8f;

#define BDIM   32
#define HDIM   256
#define WDIM   256
#define CCH    16
#define KCODES 32
#define NPIX   (BDIM*HDIM*WDIM)       // 2,097,152 pixels
#define NSTEPS 5
#define WAVES_PER_BLOCK 8
#define TILES  (NPIX/16)              // 131,072 (one wave each)
#define STEP_BLOCKS (TILES/WAVES_PER_BLOCK)

__device__ __forceinline__ v8f wmma4(v2f a, v2f b, v8f c) {
    return __builtin_amdgcn_wmma_f32_16x16x4_f32(false, a, false, b, (short)0, c, false, false);
}

// ---------------- stem: 3x3 conv 1->16 + relu, NCHW in -> NHWC state ----------------
__global__ __launch_bounds__(256) void stem_kernel(const float* __restrict__ x,
                                                   const float* __restrict__ w,   // [16][1][3][3]
                                                   const float* __restrict__ bia, // [16]
                                                   float* __restrict__ st)        // [B,H,W,16]
{
    int pix = blockIdx.x * blockDim.x + threadIdx.x;
    if (pix >= NPIX) return;
    int xx = pix % WDIM; int rest = pix / WDIM;
    int yy = rest % HDIM; int bb = rest / HDIM;
    float in9[9];
    #pragma unroll
    for (int dy = 0; dy < 3; ++dy)
        #pragma unroll
        for (int dx = 0; dx < 3; ++dx) {
            int sy = yy + dy - 1, sx = xx + dx - 1;
            bool ok = (sy >= 0) && (sy < HDIM) && (sx >= 0) && (sx < WDIM);
            float v = x[((size_t)bb*HDIM + (ok?sy:0))*WDIM + (ok?sx:0)];
            in9[dy*3+dx] = ok ? v : 0.f;
        }
    float* op = st + (size_t)pix * CCH;
    #pragma unroll
    for (int co = 0; co < CCH; ++co) {
        float a = bia[co];
        #pragma unroll
        for (int t = 0; t < 9; ++t) a = fmaf(in9[t], w[co*9 + t], a);
        op[co] = fmaxf(a, 0.f);
    }
}

// 3x3 conv accumulation over K=144 via 36 WMMAs; EDGE=false is fully unconditional.
template<bool EDGE>
__device__ __forceinline__ v8f conv3x3_acc(const float* __restrict__ sin_,
                                           int b, int y, int x0, int n, int khalf,
                                           const v2f* __restrict__ wb1)
{
    v8f acc = {0.f,0.f,0.f,0.f,0.f,0.f,0.f,0.f};
    #pragma unroll
    for (int dy = -1; dy <= 1; ++dy) {
        int yy = y + dy;
        bool yok = !EDGE || ((yy >= 0) && (yy < HDIM));
        int yc = EDGE ? (yok ? yy : 0) : yy;
        const float* rowp = sin_ + ((size_t)b*HDIM + yc) * WDIM * CCH;

        v2f av[3][4];
        #pragma unroll
        for (int dx = 0; dx < 3; ++dx) {
            int xx = x0 + n + dx - 1;
            bool ok = yok && (!EDGE || ((xx >= 0) && (xx < WDIM)));
            int xc = EDGE ? (ok ? xx : 0) : xx;
            const v2f* pv = (const v2f*)(rowp + (size_t)xc * CCH);
            float mf = ok ? 1.f : 0.f;
            #pragma unroll
            for (int g = 0; g < 4; ++g) {
                v2f t = pv[(g*4 + khalf) >> 1];       // unconditional global_load_b64
                if (EDGE) { t.x *= mf; t.y *= mf; }
                av[dx][g] = t;
            }
        }
        #pragma unroll
        for (int dx = 0; dx < 3; ++dx)
            #pragma unroll
            for (int g = 0; g < 4; ++g)
                acc = wmma4(av[dx][g], wb1[((dy+1)*3 + dx)*4 + g], acc);
    }
    return acc;
}

// ---------------- fused step: conv3x3->relu->conv1x1 + gate + VQ ----------------
__global__ __launch_bounds__(256) void step_kernel(
    const float* __restrict__ sin_, float* __restrict__ sout,
    const float* __restrict__ up1w, const float* __restrict__ up1b,
    const float* __restrict__ up2w, const float* __restrict__ up2b,
    const float* __restrict__ tauw, const float* __restrict__ taub,
    const float* __restrict__ codebook,
    float* __restrict__ loss_acc, unsigned* __restrict__ usage_mask)
{
    __shared__ __align__(16) float hbuf[WAVES_PER_BLOCK][16][16];
    __shared__ __align__(16) float cb[KCODES*CCH];
    for (int i = threadIdx.x; i < KCODES*CCH; i += 256) cb[i] = codebook[i];
    __syncthreads();

    const int lane  = threadIdx.x & 31;
    const int wv    = threadIdx.x >> 5;
    const int n     = lane & 15;            // N-col for B/C/D; pixel m for A
    const int khalf = (lane >> 4) << 1;     // 0 or 2  (A/B K pair base)
    const int off8  = (lane >> 4) << 3;     // 0 or 8  (C/D row half)

    const int tile = blockIdx.x * WAVES_PER_BLOCK + wv;
    const int tilesPerRow = WDIM / 16;
    const int x0  = (tile % tilesPerRow) * 16;
    const int row = tile / tilesPerRow;      // b*H + y
    const int y   = row % HDIM;
    const int b   = row / HDIM;
    const size_t rowBase = (size_t)row * WDIM;

    // ---- preload all B fragments into registers (hoisted out of WMMA loops) ----
    v2f wb1[36];
    #pragma unroll
    for (int t = 0; t < 9; ++t)
        #pragma unroll
        for (int g = 0; g < 4; ++g) {
            int ci = g*4 + khalf;
            const float* wp = up1w + (size_t)(n*CCH + ci)*9 + t;
            v2f w; w.x = wp[0]; w.y = wp[9];   // W[n][ci][tap], W[n][ci+1][tap]
            wb1[t*4 + g] = w;
        }
    v2f wb2[4], wbt[4];
    #pragma unroll
    for (int g = 0; g < 4; ++g) {
        int ci = g*4 + khalf;
        wb2[g] = *(const v2f*)(up2w + n*CCH + ci);
        wbt[g] = *(const v2f*)(tauw + n*CCH + ci);
    }

    // ---- h = relu(conv3x3(state, up1) + up1_b) ----
    const bool edge = (y == 0) | (y == HDIM-1) | (x0 == 0) | (x0 == WDIM-16);
    v8f acc = edge ? conv3x3_acc<true >(sin_, b, y, x0, n, khalf, wb1)
                   : conv3x3_acc<false>(sin_, b, y, x0, n, khalf, wb1);
    {
        float b1 = up1b[n];
        #pragma unroll
        for (int v = 0; v < 8; ++v) hbuf[wv][v + off8][n] = fmaxf(acc[v] + b1, 0.f);
    }
    __syncthreads();

    // ---- delta = conv1x1(h, up2); beta = sigmoid(conv1x1(state, tau)) ----
    const float* sp = sin_ + (rowBase + x0 + n) * CCH;
    v8f dacc = {0.f,0.f,0.f,0.f,0.f,0.f,0.f,0.f};
    v8f tacc = {0.f,0.f,0.f,0.f,0.f,0.f,0.f,0.f};
    #pragma unroll
    for (int g = 0; g < 4; ++g) {
        int ci = g*4 + khalf;
        v2f ah = *(const v2f*)&hbuf[wv][n][ci];          // ds_load_b64
        v2f as = *(const v2f*)(sp + ci);                 // global_load_b64
        dacc = wmma4(ah, wb2[g], dacc);
        tacc = wmma4(as, wbt[g], tacc);
    }

    // ---- gated update, D layout ----
    float b2 = up2b[n], bt = taub[n];
    float ns[8];
    #pragma unroll
    for (int v = 0; v < 8; ++v) {
        float s    = sin_[(rowBase + x0 + v + off8) * CCH + n];
        float beta = 1.f / (1.f + __expf(-(tacc[v] + bt)));
        float dlt  = dacc[v] + b2;
        ns[v] = beta * s + (1.f - beta) * dlt;
    }
    __syncthreads();                        // WAR on hbuf
    #pragma unroll
    for (int v = 0; v < 8; ++v) hbuf[wv][v + off8][n] = ns[v];
    __syncthreads();

    // ---- VQ: lanes 0..15 own one pixel each ----
    float lsum = 0.f; unsigned um = 0u;
    if (lane < 16) {
        float z[16];
        #pragma unroll
        for (int c = 0; c < 16; ++c) z[c] = hbuf[wv][n][c];
        int best = 0; float bestd = 3.4e38f;
        for (int k = 0; k < KCODES; ++k) {
            float d = 0.f;
            #pragma unroll
            for (int c = 0; c < 16; ++c) { float t = z[c] - cb[k*16 + c]; d = fmaf(t, t, d); }
            if (d < bestd) { bestd = d; best = k; }   // first-min tie rule (argmin)
        }
        float4* op = (float4*)(sout + (rowBase + x0 + n) * CCH);
        const float4* cq = (const float4*)&cb[best*16];
        op[0] = cq[0]; op[1] = cq[1]; op[2] = cq[2]; op[3] = cq[3];
        lsum = bestd;                       // == sum_c (z_q - z)^2
        um = 1u << best;
    }
    #pragma unroll
    for (int s = 16; s >= 1; s >>= 1) {
        lsum += __shfl_xor(lsum, s, 32);
        um   |= (unsigned)__shfl_xor((int)um, s, 32);
    }
    if (lane == 0) { atomicAdd(loss_acc, lsum); atomicOr(usage_mask, um); }
}

// ---------------- decode: sigmoid(conv1x1 16->1) ----------------
__global__ __launch_bounds__(256) void decode_kernel(const float* __restrict__ st,
                                                     const float* __restrict__ dw,
                                                     const float* __restrict__ db,
                                                     float* __restrict__ out)
{
    int pix = blockIdx.x * blockDim.x + threadIdx.x;
    if (pix >= NPIX) return;
    const float* sp = st + (size_t)pix * CCH;
    float a = db[0];
    #pragma unroll
    for (int c = 0; c < CCH; ++c) a = fmaf(sp[c], dw[c], a);
    out[pix] = 1.f / (1.f + __expf(-a));
}

__global__ void init_acc(float* loss, unsigned* mask) {
    int i = threadIdx.x;
    if (i < NSTEPS) { loss[i] = 0.f; mask[i] = 0u; }
}

__global__ void finalize_kernel(const float* loss, const unsigned* mask, float* out2) {
    if (threadIdx.x == 0) {
        float vt = 0.f, ut = 0.f;
        for (int i = 0; i < NSTEPS; ++i) {
            vt += 1.25f * loss[i] / (float)((size_t)NPIX * CCH); // codebook + 0.25*commit
            ut += (float)__popc(mask[i]) / (float)KCODES;
        }
        out2[0] = vt / (float)NSTEPS;
        out2[1] = ut / (float)NSTEPS;
    }
}

extern "C" void kernel_launch(void* const* d_in, const int* in_sizes, int n_in,
                              void* d_out, int out_size, void* d_ws, size_t ws_size,
                              hipStream_t stream)
{
    const float* x        = (const float*)d_in[0];
    const float* stem_w   = (const float*)d_in[1];
    const float* stem_b   = (const float*)d_in[2];
    const float* up1_w    = (const float*)d_in[3];
    const float* up1_b    = (const float*)d_in[4];
    const float* up2_w    = (const float*)d_in[5];
    const float* up2_b    = (const float*)d_in[6];
    const float* tau_w    = (const float*)d_in[7];
    const float* tau_b    = (const float*)d_in[8];
    const float* codebook = (const float*)d_in[9];
    const float* dec_w    = (const float*)d_in[10];
    const float* dec_b    = (const float*)d_in[11];
    // d_in[12] = n_steps (device scalar); fixed at 5 per reference.

    float* out = (float*)d_out;
    char* ws = (char*)d_ws;
    const size_t stateBytes = (size_t)NPIX * CCH * sizeof(float);
    float*    stateA     = (float*)ws;
    float*    stateB     = (float*)(ws + stateBytes);
    float*    loss_acc   = (float*)(ws + 2*stateBytes);
    unsigned* usage_mask = (unsigned*)(loss_acc + NSTEPS);

    init_acc<<<1, 64, 0, stream>>>(loss_acc, usage_mask);
    stem_kernel<<<NPIX/256, 256, 0, stream>>>(x, stem_w, stem_b, stateA);

    float* sIn = stateA; float* sOut = stateB;
    for (int s = 0; s < NSTEPS; ++s) {
        step_kernel<<<STEP_BLOCKS, 256, 0, stream>>>(sIn, sOut,
            up1_w, up1_b, up2_w, up2_b, tau_w, tau_b, codebook,
            loss_acc + s, usage_mask + s);
        float* t = sIn; sIn = sOut; sOut = t;
    }

    decode_kernel<<<NPIX/256, 256, 0, stream>>>(sIn, dec_w, dec_b, out);
    finalize_kernel<<<1, 32, 0, stream>>>(loss_acc, usage_mask, out + NPIX);
}